// TrmEncoder_19859928776705
// MI455X (gfx1250) — compile-verified
//
#include <hip/hip_runtime.h>
#include <math.h>

// CDNA5 WMMA fp32 fragments (wave32): A=16x4 -> 2 VGPRs, B=4x16 -> 2 VGPRs, C/D=16x16 -> 8 VGPRs
typedef float f32x2 __attribute__((ext_vector_type(2)));
typedef float f32x8 __attribute__((ext_vector_type(8)));

__device__ __forceinline__ f32x8 wmma_f32(f32x2 a, f32x2 b, f32x8 c) {
    // (neg_a, A, neg_b, B, c_mod, C, reuse_a, reuse_b)
    return __builtin_amdgcn_wmma_f32_16x16x4_f32(false, a, false, b, (short)0, c, false, false);
}

// gfx1250 async memory->LDS copy (ASYNCcnt-tracked, no VGPR round trip).
// LDS operand is the low 32 bits of the generic pointer (LDS aperture keeps
// the byte offset in addr[31:0]).
__device__ __forceinline__ void async_cp_b128(void* lds, const void* gptr) {
    asm volatile("global_load_async_to_lds_b128 %0, %1, off"
                 :: "v"((unsigned)(uintptr_t)lds),
                    "v"((unsigned long long)(uintptr_t)gptr)
                 : "memory");
}
__device__ __forceinline__ void wait_async0() {
    asm volatile("s_wait_asynccnt 0" ::: "memory");
}

// ---------------------------------------------------------------------------
// GEMM: C[M,N] = A[M,K] @ W[N,K]^T (+bias) (+ReLU)
// 256 threads = 8 waves, block tile 64x64, double-buffered async K staging.
// requires M%64==0, N%64==0, K%32==0 (true for all uses here)
// ---------------------------------------------------------------------------
#define GEMM_KC 32

__global__ __launch_bounds__(256) void trm_gemm_wmma(
    const float* __restrict__ A, const float* __restrict__ W,
    const float* __restrict__ bias, float* __restrict__ C,
    int M, int N, int K, int relu)
{
    __shared__ float As[2][64][GEMM_KC + 4];
    __shared__ float Ws[2][64][GEMM_KC + 4];

    const int tid  = threadIdx.x;
    const int wave = tid >> 5;
    const int lane = tid & 31;
    const int lh   = lane >> 4;     // lane-half: selects K pair / M+8 rows
    const int l16  = lane & 15;
    const int wm   = wave >> 1;     // 0..3 -> 16-row strip
    const int wn   = wave & 1;      // 0..1 -> 32-col strip
    const int m0   = blockIdx.y * 64;
    const int n0   = blockIdx.x * 64;

    const int ldr = tid >> 3;          // 0..31 (x2 -> 0..63)
    const int ldc = (tid & 7) << 2;    // 0,4,...,28

    f32x8 acc0 = {};
    f32x8 acc1 = {};

    // prologue: async-stage K chunk 0 into buffer 0
#pragma unroll
    for (int i = 0; i < 2; ++i) {
        const int r = ldr + i * 32;
        async_cp_b128(&As[0][r][ldc], A + (size_t)(m0 + r) * K + ldc);
        async_cp_b128(&Ws[0][r][ldc], W + (size_t)(n0 + r) * K + ldc);
    }

    int buf = 0;
    for (int k0 = 0; k0 < K; k0 += GEMM_KC, buf ^= 1) {
        wait_async0();          // my async writes for this stage have landed
        __syncthreads();        // everyone's landed; prev compute done reading buf^1

        if (k0 + GEMM_KC < K) { // issue next stage into other buffer; overlaps WMMAs below
#pragma unroll
            for (int i = 0; i < 2; ++i) {
                const int r = ldr + i * 32;
                async_cp_b128(&As[buf ^ 1][r][ldc],
                              A + (size_t)(m0 + r) * K + k0 + GEMM_KC + ldc);
                async_cp_b128(&Ws[buf ^ 1][r][ldc],
                              W + (size_t)(n0 + r) * K + k0 + GEMM_KC + ldc);
            }
        }
        if (k0 + 2 * GEMM_KC < K) {  // warm L2 two stages ahead -> global_prefetch_b8
            __builtin_prefetch(A + (size_t)(m0 + ldr) * K + k0 + 2 * GEMM_KC + ldc, 0, 0);
            __builtin_prefetch(W + (size_t)(n0 + ldr) * K + k0 + 2 * GEMM_KC + ldc, 0, 0);
        }

#pragma unroll
        for (int kk = 0; kk < GEMM_KC; kk += 4) {
            f32x2 a;
            a.x = As[buf][wm * 16 + l16][kk + 2 * lh];
            a.y = As[buf][wm * 16 + l16][kk + 2 * lh + 1];
            f32x2 b0;
            b0.x = Ws[buf][wn * 32 + l16][kk + 2 * lh];
            b0.y = Ws[buf][wn * 32 + l16][kk + 2 * lh + 1];
            f32x2 b1;
            b1.x = Ws[buf][wn * 32 + 16 + l16][kk + 2 * lh];
            b1.y = Ws[buf][wn * 32 + 16 + l16][kk + 2 * lh + 1];
            acc0 = wmma_f32(a, b0, acc0);
            acc1 = wmma_f32(a, b1, acc1);
        }
    }

    const int col0 = n0 + wn * 32 + l16;
    const int col1 = col0 + 16;
    const float bv0 = bias ? bias[col0] : 0.0f;
    const float bv1 = bias ? bias[col1] : 0.0f;
#pragma unroll
    for (int r = 0; r < 8; ++r) {
        const int row = m0 + wm * 16 + r + 8 * lh;  // C layout: M = r + 8*half
        float v0 = acc0[r] + bv0;
        float v1 = acc1[r] + bv1;
        if (relu) { v0 = fmaxf(v0, 0.0f); v1 = fmaxf(v1, 0.0f); }
        C[(size_t)row * N + col0] = v0;
        C[(size_t)row * N + col1] = v1;
    }
}

// ---------------------------------------------------------------------------
// Flash attention (no 1/sqrt(hd) scaling, matching reference).
// grid = (S/128, B*H); block = 256 = 8 waves; wave owns 16 query rows.
// Q is held in registers (each lane keeps exactly the 32 floats its A-fragment
// slots need); K/V tiles are double-buffered in LDS via async copies.
// qkv: [S][B][3D], out: [S][B][D], hd = 64.
// ---------------------------------------------------------------------------
#define HD 64
#define QROWS 128
#define KT 16

__global__ __launch_bounds__(256) void trm_attn_flash(
    const float* __restrict__ qkv, float* __restrict__ out,
    int S, int B, int H)
{
    const int D    = H * HD;
    const int nh   = blockIdx.y;        // b*H + h
    const int b    = nh / H;
    const int h    = nh % H;
    const int q0   = blockIdx.x * QROWS;
    const int tid  = threadIdx.x;
    const int wave = tid >> 5;
    const int lane = tid & 31;
    const int lh   = lane >> 4;
    const int l16  = lane & 15;
    const int row3D = B * 3 * D;

    __shared__ float Ks[2][KT][HD + 4];
    __shared__ float Vs[2][KT][HD + 4];
    __shared__ float Ps[8][16][17];     // per-wave C-layout -> A-layout bounce

    const float* qbase = qkv + (size_t)b * 3 * D + h * HD;
    const float* kbase = qbase + D;
    const float* vbase = qbase + 2 * D;

    // Q fragment slots in registers: lane (lh, l16) of wave w supplies
    // A-elements (M = l16, K = 4c + 2*lh + {0,1}) for c = 0..15.
    const int qrow = q0 + wave * 16 + l16;
    float qv[32];
#pragma unroll
    for (int c = 0; c < 16; ++c) {
        const float2 t = *(const float2*)(qbase + (size_t)qrow * row3D + 4 * c + 2 * lh);
        qv[2 * c]     = t.x;
        qv[2 * c + 1] = t.y;
    }

    const int skr = tid >> 4;           // 0..15
    const int skc = (tid & 15) << 2;    // 0,4,...,60

    // prologue: async-stage key tile 0
    async_cp_b128(&Ks[0][skr][skc], kbase + (size_t)skr * row3D + skc);
    async_cp_b128(&Vs[0][skr][skc], vbase + (size_t)skr * row3D + skc);

    float m_row[8], l_row[8];
    f32x8 o[4] = {};
#pragma unroll
    for (int r = 0; r < 8; ++r) { m_row[r] = -3.0e38f; l_row[r] = 0.0f; }

    int buf = 0;
    for (int kt = 0; kt < S; kt += KT, buf ^= 1) {
        wait_async0();
        __syncthreads();

        if (kt + KT < S) {  // stage next key tile; overlaps the WMMAs below
            async_cp_b128(&Ks[buf ^ 1][skr][skc],
                          kbase + (size_t)(kt + KT + skr) * row3D + skc);
            async_cp_b128(&Vs[buf ^ 1][skr][skc],
                          vbase + (size_t)(kt + KT + skr) * row3D + skc);
        }

        // S_tile(16x16) = Q(16x64) @ K^T(64x16), 16 fp32 WMMA steps (A from regs)
        f32x8 s = {};
#pragma unroll
        for (int c = 0; c < 16; ++c) {
            f32x2 a;
            a.x = qv[2 * c];
            a.y = qv[2 * c + 1];
            f32x2 bb;
            bb.x = Ks[buf][l16][4 * c + 2 * lh];
            bb.y = Ks[buf][l16][4 * c + 2 * lh + 1];
            s = wmma_f32(a, bb, s);
        }

        // online softmax; C row m = r + 8*lh lives on the 16 lanes of this half,
        // so xor masks {1,2,4,8} reduce exactly over one row.
#pragma unroll
        for (int r = 0; r < 8; ++r) {
            float sv = s[r];
            float rmax = sv;
            rmax = fmaxf(rmax, __shfl_xor(rmax, 1, 32));
            rmax = fmaxf(rmax, __shfl_xor(rmax, 2, 32));
            rmax = fmaxf(rmax, __shfl_xor(rmax, 4, 32));
            rmax = fmaxf(rmax, __shfl_xor(rmax, 8, 32));
            const float mnew = fmaxf(m_row[r], rmax);
            const float p = __expf(sv - mnew);
            float rsum = p;
            rsum += __shfl_xor(rsum, 1, 32);
            rsum += __shfl_xor(rsum, 2, 32);
            rsum += __shfl_xor(rsum, 4, 32);
            rsum += __shfl_xor(rsum, 8, 32);
            const float alpha = __expf(m_row[r] - mnew);
            l_row[r] = l_row[r] * alpha + rsum;
            m_row[r] = mnew;
#pragma unroll
            for (int nt = 0; nt < 4; ++nt) o[nt][r] *= alpha;
            Ps[wave][r + 8 * lh][l16] = p;   // per-wave scratch, same-wave LDS is in-order
        }

        // O(16x64) += P(16x16) @ V(16x64): 4 K-chunks x 4 N-tiles
#pragma unroll
        for (int kk = 0; kk < 16; kk += 4) {
            f32x2 a;
            a.x = Ps[wave][l16][kk + 2 * lh];
            a.y = Ps[wave][l16][kk + 2 * lh + 1];
#pragma unroll
            for (int nt = 0; nt < 4; ++nt) {
                f32x2 bb;
                bb.x = Vs[buf][kk + 2 * lh][nt * 16 + l16];
                bb.y = Vs[buf][kk + 2 * lh + 1][nt * 16 + l16];
                o[nt] = wmma_f32(a, bb, o[nt]);
            }
        }
    }

    // normalize and write: out[s][b][h*64 + n]
#pragma unroll
    for (int r = 0; r < 8; ++r) {
        const float inv = 1.0f / l_row[r];
        const int row = q0 + wave * 16 + r + 8 * lh;
#pragma unroll
        for (int nt = 0; nt < 4; ++nt) {
            out[(size_t)row * (B * D) + b * D + h * HD + nt * 16 + l16] = o[nt][r] * inv;
        }
    }
}

// ---------------------------------------------------------------------------
// Fused residual + LayerNorm over D=512. One block (256 thr) per row.
// ---------------------------------------------------------------------------
__global__ __launch_bounds__(256) void trm_add_layernorm(
    const float* __restrict__ x, const float* __restrict__ y,
    const float* __restrict__ gamma, const float* __restrict__ beta,
    float* __restrict__ outp, int D)
{
    const int row = blockIdx.x;
    const int tid = threadIdx.x;
    const float* xr = x + (size_t)row * D;
    const float* yr = y + (size_t)row * D;

    const float v0 = xr[tid] + yr[tid];
    const float v1 = xr[tid + 256] + yr[tid + 256];

    float s  = v0 + v1;
    float s2 = v0 * v0 + v1 * v1;
#pragma unroll
    for (int off = 16; off >= 1; off >>= 1) {
        s  += __shfl_xor(s,  off, 32);
        s2 += __shfl_xor(s2, off, 32);
    }
    __shared__ float red[2][8];
    const int wave = tid >> 5, lane = tid & 31;
    if (lane == 0) { red[0][wave] = s; red[1][wave] = s2; }
    __syncthreads();
    s = 0.0f; s2 = 0.0f;
#pragma unroll
    for (int i = 0; i < 8; ++i) { s += red[0][i]; s2 += red[1][i]; }

    const float mu  = s / (float)D;
    const float var = s2 / (float)D - mu * mu;
    const float inv = rsqrtf(var + 1e-5f);

    outp[(size_t)row * D + tid]       = (v0 - mu) * inv * gamma[tid]       + beta[tid];
    outp[(size_t)row * D + tid + 256] = (v1 - mu) * inv * gamma[tid + 256] + beta[tid + 256];
}

// ---------------------------------------------------------------------------
extern "C" void kernel_launch(void* const* d_in, const int* in_sizes, int n_in,
                              void* d_out, int out_size, void* d_ws, size_t ws_size,
                              hipStream_t stream) {
    const float* src   = (const float*)d_in[0];   // [S,B,D]
    const float* qkv_w = (const float*)d_in[1];   // [3D,D]
    const float* qkv_b = (const float*)d_in[2];   // [3D]
    const float* out_w = (const float*)d_in[3];   // [D,D]
    const float* w1    = (const float*)d_in[4];   // [F,D]
    const float* b1    = (const float*)d_in[5];   // [F]
    const float* w2    = (const float*)d_in[6];   // [D,F]
    const float* b2    = (const float*)d_in[7];   // [D]
    const float* ln1_g = (const float*)d_in[8];
    const float* ln1_b = (const float*)d_in[9];
    const float* ln2_g = (const float*)d_in[10];
    const float* ln2_b = (const float*)d_in[11];

    const int S = 4096, B = 2, D = 512, H = 8, F = 2048;
    const int M = S * B;  // 8192 rows

    // Workspace layout (aliased; peak = 29,360,128 floats = 112 MiB)
    float* R0 = (float*)d_ws;                       // M*F (16.8M floats): qkv (12.6M) then ffn_h
    float* R1 = R0 + (size_t)M * F;                 // M*D: attn_out, then ffn_y
    float* R2 = R1 + (size_t)M * D;                 // M*D: attn proj
    float* R3 = R2 + (size_t)M * D;                 // M*D: x1 (post-LN1)

    float* qkv   = R0;
    float* attnO = R1;
    float* proj  = R2;
    float* x1    = R3;
    float* ffh   = R0;   // qkv dead after attention
    float* ffy   = R1;   // attnO dead after out-proj

    // 1) qkv = src @ qkv_w^T + qkv_b          [8192, 1536]
    trm_gemm_wmma<<<dim3(3 * D / 64, M / 64), 256, 0, stream>>>(
        src, qkv_w, qkv_b, qkv, M, 3 * D, D, 0);

    // 2) flash attention -> attnO             [8192, 512]
    trm_attn_flash<<<dim3(S / QROWS, B * H), 256, 0, stream>>>(qkv, attnO, S, B, H);

    // 3) proj = attnO @ out_w^T (no bias)     [8192, 512]
    trm_gemm_wmma<<<dim3(D / 64, M / 64), 256, 0, stream>>>(
        attnO, out_w, nullptr, proj, M, D, D, 0);

    // 4) x1 = LN(src + proj)
    trm_add_layernorm<<<M, 256, 0, stream>>>(src, proj, ln1_g, ln1_b, x1, D);

    // 5) ffh = relu(x1 @ w1^T + b1)           [8192, 2048]
    trm_gemm_wmma<<<dim3(F / 64, M / 64), 256, 0, stream>>>(
        x1, w1, b1, ffh, M, F, D, 1);

    // 6) ffy = ffh @ w2^T + b2                [8192, 512]
    trm_gemm_wmma<<<dim3(D / 64, M / 64), 256, 0, stream>>>(
        ffh, w2, b2, ffy, M, D, F, 0);

    // 7) out = LN(x1 + ffy)
    trm_add_layernorm<<<M, 256, 0, stream>>>(x1, ffy, ln2_g, ln2_b, (float*)d_out, D);
}